// GAT_Decoder_56788057588259
// MI455X (gfx1250) — compile-verified
//
#include <hip/hip_runtime.h>
#include <hip/hip_bf16.h>
#include <math.h>

typedef __attribute__((ext_vector_type(2))) float v2f;
typedef __attribute__((ext_vector_type(8))) float v8f;

#define NN 128
#define HH 128
#define NHEAD 8
#define DHEAD 16

// ---------------------------------------------------------------------------
// One 16x16 output tile of  D = A(16xK) * W^T(Kx16)  using V_WMMA_F32_16X16X4_F32.
// A is row-major with leading dim lda.  Wrow points at W + n0*ldw (row-major,
// W is [N][K] so B[k][n] = W[n][k]; lanes index n, lane-half selects K pairs).
// ---------------------------------------------------------------------------
__device__ __forceinline__ v8f wmma_tile_f32(const float* __restrict__ A, int lda,
                                             const float* __restrict__ Wrow, int ldw,
                                             int K, int lane) {
    const int half = lane >> 4;
    const int l    = lane & 15;
    v8f c = {0.f, 0.f, 0.f, 0.f, 0.f, 0.f, 0.f, 0.f};
    for (int k0 = 0; k0 < K; k0 += 4) {
        v2f a, b;
        const int ka = k0 + 2 * half;
        a.x = A[l * lda + ka];
        a.y = A[l * lda + ka + 1];
        b.x = Wrow[l * ldw + ka];
        b.y = Wrow[l * ldw + ka + 1];
        c = __builtin_amdgcn_wmma_f32_16x16x4_f32(false, a, false, b, (short)0, c,
                                                  false, false);
    }
    return c;
}

// C/D layout: VGPR r -> row (r + 8*half), col = lane&15
__device__ __forceinline__ void store_tile(float* dst, int ldd, int n0, v8f c, int lane) {
    const int half = lane >> 4;
    const int l    = lane & 15;
    float* o = dst + (8 * half) * ldd + n0 + l;
#pragma unroll
    for (int r = 0; r < 8; ++r) o[r * ldd] = c[r];
}

// ---------------------------------------------------------------------------
// Precompute Kc = enc @ Wm_k^T, Vc = enc @ Wm_v^T, Kp = enc @ Wp_k^T
// enc viewed as (B*N, H) row-major.  One wave per 16x16 output tile.
// ---------------------------------------------------------------------------
__global__ __launch_bounds__(256) void gat_precompute_kernel(
    const float* __restrict__ enc,
    const float* __restrict__ Wm_k, const float* __restrict__ Wm_v,
    const float* __restrict__ Wp_k,
    float* __restrict__ Kc, float* __restrict__ Vc, float* __restrict__ Kp,
    int mtiles) {
    const int lane = threadIdx.x & 31;
    const int wave = blockIdx.x * (blockDim.x >> 5) + (threadIdx.x >> 5);
    const int per_mat = mtiles * 8;
    const int mat   = wave / per_mat;
    const int rem   = wave - mat * per_mat;
    const int mtile = rem >> 3;
    const int ntile = rem & 7;
    if (mat >= 3) return;

    const float* W  = (mat == 0) ? Wm_k : (mat == 1) ? Wm_v : Wp_k;
    float*      Out = (mat == 0) ? Kc   : (mat == 1) ? Vc   : Kp;

    const float* A = enc + (size_t)mtile * 16 * HH;
    v8f c = wmma_tile_f32(A, HH, W + ntile * 16 * HH, HH, HH, lane);
    store_tile(Out + (size_t)mtile * 16 * HH, HH, ntile * 16, c, lane);
}

// ---------------------------------------------------------------------------
// Persistent decode kernel: each block owns 16 batch rows, runs all n_steps
// sequentially with state in LDS.  8 waves; wave w owns output columns
// [16w, 16w+16) in each WMMA GEMM.
// ---------------------------------------------------------------------------
__global__ __launch_bounds__(256) void gat_decode_kernel(
    const float* __restrict__ enc, const float* __restrict__ pool,
    const float* __restrict__ capacity, const float* __restrict__ demand,
    const float* __restrict__ W_fc, const float* __restrict__ W_fc1,
    const float* __restrict__ Wm_w, const float* __restrict__ Wm_fc,
    const float* __restrict__ Kc, const float* __restrict__ Vc,
    const float* __restrict__ Kp,
    const int* __restrict__ T_p,
    float* __restrict__ out_actions, float* __restrict__ out_logps,
    int n_steps) {
    extern __shared__ float smem[];
    float* s_input  = smem;            // 16*128
    float* s_pool   = smem + 2048;     // 16*128
    float* s_dem    = smem + 4096;     // 16*128
    float* s_dec    = smem + 6144;     // 16*128
    float* s_q      = smem + 8192;     // 16*128
    float* s_out    = smem + 10240;    // 16*128
    float* s_x      = smem + 12288;    // 16*128
    float* s_logit  = smem + 14336;    // 16*128
    float* s_mask   = smem + 16384;    // 16*128
    float* s_mask1  = smem + 18432;    // 16*128
    float* s_compat = smem + 20480;    // 16*8*128
    float* s_cap    = smem + 36864;    // 16
    float* s_lps    = smem + 36880;    // 16
    int*   s_idx    = (int*)(smem + 36896); // 16

    const int tid  = threadIdx.x;
    const int lane = tid & 31;
    const int wave = tid >> 5;
    const int bg0  = blockIdx.x * 16;

    const float Tf       = (float)T_p[0];
    const float max_cap  = capacity[0];
    const float mha_norm = 0.25f;                 // 1/sqrt(16)
    const float ptr_norm = 0.08838834764831845f;  // 1/sqrt(128)
    const float NEG_INF  = -__builtin_inff();

    // ---- init state ----
    for (int p = tid; p < 16 * NN; p += 256) {
        const int b = p >> 7, n = p & 127;
        s_input[p] = enc[((size_t)(bg0 + b) * NN + 0) * HH + n];
        s_dem[p]   = demand[(size_t)(bg0 + b) * NN + n];
        s_mask[p]  = 0.f;
        s_mask1[p] = 0.f;
    }
    if (tid < 16) {
        s_cap[tid] = capacity[bg0 + tid];
        s_lps[tid] = 0.f;
    }
    __syncthreads();

    // pool_proc = pool @ W_fc1^T  (WMMA, one 16-col tile per wave)
    {
        v8f c = wmma_tile_f32(pool + (size_t)bg0 * HH, HH,
                              W_fc1 + wave * 16 * HH, HH, HH, lane);
        store_tile(s_pool, NN, wave * 16, c, lane);
    }
    __syncthreads();

    for (int i = 0; i < n_steps; ++i) {
        // ---- step-0 mask init: mask = update_mask(cap, 0, mask1, false) ----
        if (i == 0) {
            for (int p = tid; p < 16 * NN; p += 256) {
                const int b = p >> 7;
                s_mask[p] = fmaxf(s_mask1[p], (s_dem[p] > s_cap[b]) ? 1.f : 0.f);
            }
            __syncthreads();
            if (tid < 16) {
                bool all = true;
                for (int n = 1; n < NN; ++n) all = all && (s_mask[tid * NN + n] > 0.f);
                if (all) s_mask[tid * NN] = 0.f;
            }
            __syncthreads();
        }

        // ---- dec = concat(input, cap) @ W_fc^T + pool_proc  (WMMA + rank-1) ----
        {
            const int half = lane >> 4;
            const int l    = lane & 15;
            v8f c = {0.f, 0.f, 0.f, 0.f, 0.f, 0.f, 0.f, 0.f};
            const float* Wr = W_fc + (size_t)(wave * 16) * (HH + 1);
            for (int k0 = 0; k0 < HH; k0 += 4) {
                v2f a, b;
                const int ka = k0 + 2 * half;
                a.x = s_input[l * NN + ka];
                a.y = s_input[l * NN + ka + 1];
                b.x = Wr[l * (HH + 1) + ka];
                b.y = Wr[l * (HH + 1) + ka + 1];
                c = __builtin_amdgcn_wmma_f32_16x16x4_f32(false, a, false, b,
                                                          (short)0, c, false, false);
            }
#pragma unroll
            for (int r = 0; r < 8; ++r) {
                const int row = 8 * half + r;
                const int col = wave * 16 + l;
                s_dec[row * NN + col] =
                    c[r] + s_cap[row] * W_fc[(size_t)col * (HH + 1) + HH] +
                    s_pool[row * NN + col];
            }
        }
        __syncthreads();

        // ---- q = dec @ Wm_w^T ----
        {
            v8f c = wmma_tile_f32(s_dec, NN, Wm_w + wave * 16 * HH, HH, HH, lane);
            store_tile(s_q, NN, wave * 16, c, lane);
        }
        __syncthreads();

        // ---- compat[b,h,n] = mha_norm * q[b,h,:].Kc[b,n,h,:]  (masked) ----
        for (int p = tid; p < 16 * NN; p += 256) {
            const int b = p >> 7, n = p & 127;
            const bool masked = s_mask[p] > 0.f;
            const float* kcp = Kc + ((size_t)(bg0 + b) * NN + n) * HH;
            const float* qb  = s_q + b * NN;
#pragma unroll
            for (int h = 0; h < NHEAD; ++h) {
                float acc = 0.f;
#pragma unroll
                for (int d = 0; d < DHEAD; ++d)
                    acc += qb[h * DHEAD + d] * kcp[h * DHEAD + d];
                s_compat[(b * NHEAD + h) * NN + n] =
                    masked ? NEG_INF : acc * mha_norm;
            }
        }
        __syncthreads();

        // ---- softmax over n, per (b,h) ----
        if (tid < 16 * NHEAD) {
            float* row = s_compat + tid * NN;
            float m = NEG_INF;
            for (int n = 0; n < NN; ++n) m = fmaxf(m, row[n]);
            if (m == NEG_INF) {
                for (int n = 0; n < NN; ++n) row[n] = 0.f;
            } else {
                float s = 0.f;
                for (int n = 0; n < NN; ++n) {
                    const float e = __expf(row[n] - m);
                    row[n] = e;
                    s += e;
                }
                const float inv = 1.f / s;
                for (int n = 0; n < NN; ++n) row[n] *= inv;
            }
        }
        __syncthreads();

        // ---- out[b, h*16+d] = sum_n attn[b,h,n] * Vc[b,n,h*16+d] ----
        for (int p = tid; p < 16 * HH; p += 256) {
            const int b = p >> 7, c = p & 127;
            const int h = c >> 4;
            const float* attn = s_compat + (b * NHEAD + h) * NN;
            const float* vcp  = Vc + (size_t)(bg0 + b) * NN * HH + c;
            float acc = 0.f;
            for (int n = 0; n < NN; ++n) acc += attn[n] * vcp[n * HH];
            s_out[p] = acc;
        }
        __syncthreads();

        // ---- x = out @ Wm_fc^T ----
        {
            v8f c = wmma_tile_f32(s_out, NN, Wm_fc + wave * 16 * HH, HH, HH, lane);
            store_tile(s_x, NN, wave * 16, c, lane);
        }
        __syncthreads();

        // ---- logits[b,n] = mask ? -inf : 10*tanh(ptr_norm * x[b].Kp[b,n]) ----
        for (int p = tid; p < 16 * NN; p += 256) {
            const int b = p >> 7, n = p & 127;
            const float* kpp = Kp + ((size_t)(bg0 + b) * NN + n) * HH;
            const float* xb  = s_x + b * NN;
            float acc = 0.f;
#pragma unroll 4
            for (int c = 0; c < HH; ++c) acc += xb[c] * kpp[c];
            const float lg = tanhf(acc * ptr_norm) * 10.f;
            s_logit[p] = (s_mask[p] > 0.f) ? NEG_INF : lg;
        }
        __syncthreads();

        // ---- per-batch-row serial part: argmax, log-softmax, cap update ----
        if (tid < 16) {
            const int b = tid;
            const float* lg = s_logit + b * NN;
            float best = NEG_INF;
            int   bi   = 0;
            for (int n = 0; n < NN; ++n)
                if (lg[n] > best) { best = lg[n]; bi = n; }
            const float m2 = best / Tf;  // max of logits/T
            float s = 0.f;
            for (int n = 0; n < NN; ++n) s += __expf(lg[n] / Tf - m2);
            float lp = best / Tf - (m2 + __logf(s));
            float cnt = 0.f;
            for (int n = 1; n < NN; ++n) cnt += s_mask1[b * NN + n];
            if (cnt >= (float)(NN - 1)) lp = 0.f;
            s_lps[b] += lp;
            s_idx[b] = bi;
            s_cap[b] = (bi == 0) ? max_cap : s_cap[b] - s_dem[b * NN + bi];
            out_actions[(size_t)(bg0 + b) * n_steps + i] = (float)bi;
        }
        __syncthreads();

        // ---- update_mask(cap, idx, mask1, i>0) + fetch next input row ----
        for (int p = tid; p < 16 * NN; p += 256) {
            const int b  = p >> 7, n = p & 127;
            const int bi = s_idx[b];
            float m1 = s_mask1[p];
            if (i > 0 && n == bi) m1 = 1.f;
            s_mask1[p] = m1;
            float m = fmaxf(m1, (s_dem[p] > s_cap[b]) ? 1.f : 0.f);
            if (i > 0 && n == bi) m = 1.f;
            s_mask[p] = m;
            s_input[p] = enc[((size_t)(bg0 + b) * NN + bi) * HH + n];
        }
        __syncthreads();
        if (tid < 16) {
            bool all = true;
            for (int n = 1; n < NN; ++n) all = all && (s_mask[tid * NN + n] > 0.f);
            if (all) s_mask[tid * NN] = 0.f;
        }
        __syncthreads();
    }

    if (tid < 16) out_logps[bg0 + tid] = s_lps[tid];
}

// ---------------------------------------------------------------------------
extern "C" void kernel_launch(void* const* d_in, const int* in_sizes, int n_in,
                              void* d_out, int out_size, void* d_ws, size_t ws_size,
                              hipStream_t stream) {
    const float* enc      = (const float*)d_in[0];
    const float* pool     = (const float*)d_in[1];
    const float* capacity = (const float*)d_in[2];
    const float* demand   = (const float*)d_in[3];
    const float* W_fc     = (const float*)d_in[4];
    const float* W_fc1    = (const float*)d_in[5];
    const float* Wp_k     = (const float*)d_in[6];
    const float* Wm_w     = (const float*)d_in[7];
    const float* Wm_k     = (const float*)d_in[8];
    const float* Wm_v     = (const float*)d_in[9];
    const float* Wm_fc    = (const float*)d_in[10];
    const int*   T_p      = (const int*)d_in[12];

    const int B  = in_sizes[1] / HH;      // pool is (B, H)
    const int ns = out_size / B - 1;      // out = B*ns actions + B logps

    float* Kc = (float*)d_ws;
    float* Vc = Kc + (size_t)B * NN * HH;
    float* Kp = Vc + (size_t)B * NN * HH;

    // precompute Kc/Vc/Kp: one wave per 16x16 tile
    const int mtiles = (B * NN) / 16;
    const int waves  = 3 * mtiles * 8;
    const int blocks = (waves + 7) / 8;
    gat_precompute_kernel<<<blocks, 256, 0, stream>>>(enc, Wm_k, Wm_v, Wp_k,
                                                      Kc, Vc, Kp, mtiles);

    float* out_actions = (float*)d_out;
    float* out_logps   = out_actions + (size_t)B * ns;
    const size_t smem_bytes = (size_t)(36912 + 16) * sizeof(float);
    gat_decode_kernel<<<B / 16, 256, smem_bytes, stream>>>(
        enc, pool, capacity, demand, W_fc, W_fc1, Wm_w, Wm_fc, Kc, Vc, Kp,
        T_p, out_actions, out_logps, ns);
}